// SpikingGCNConv_77747497992593
// MI455X (gfx1250) — compile-verified
//
#include <hip/hip_runtime.h>
#include <hip/hip_bf16.h>

#define N_NODES 50000
#define IN_CH   256
#define OUT_CH  128
#define N_EDGES 800000
#define BETA    0.9f
#define THRESH  1.0f

typedef float v2f __attribute__((ext_vector_type(2)));
typedef float v8f __attribute__((ext_vector_type(8)));

// ---------------------------------------------------------------------------
// 1) degree init: deg[i] = 1.0 (self-loop)
__global__ void deg_init_kernel(float* __restrict__ deg) {
    int i = blockIdx.x * blockDim.x + threadIdx.x;
    if (i < N_NODES) deg[i] = 1.0f;
}

// 2) degree accumulation over edge destinations
__global__ void deg_accum_kernel(const int* __restrict__ edge_index,
                                 float* __restrict__ deg) {
    int e = blockIdx.x * blockDim.x + threadIdx.x;
    if (e < N_EDGES) {
        int d = edge_index[N_EDGES + e];  // edge_index[1][e]
        __hip_atomic_fetch_add(&deg[d], 1.0f,
                               __ATOMIC_RELAXED, __HIP_MEMORY_SCOPE_AGENT);
    }
}

// 3) dinv[i] = rsqrt(deg[i])   (deg >= 1 always due to self-loop)
__global__ void deg_rsqrt_kernel(float* __restrict__ deg) {
    int i = blockIdx.x * blockDim.x + threadIdx.x;
    if (i < N_NODES) deg[i] = rsqrtf(deg[i]);
}

// ---------------------------------------------------------------------------
// 4) GEMM h = x @ W  via V_WMMA_F32_16X16X4_F32 (full fp32, exact)
//    block = 256 threads = 8 waves; wave w -> N-tile w; block m -> 16 rows.
//    A 16x4 f32 layout: lane = {M = lane&15}, VGPR v holds K = kk + 2*(lane>>4) + v
//    B 4x16  layout:    lane = {N = lane&15}, VGPR v holds K = kk + 2*(lane>>4) + v
//    C/D 16x16: VGPR r, lanes0-15 -> (M=r, N=lane), lanes16-31 -> (M=r+8, N=lane-16)
__global__ void __launch_bounds__(256)
gemm_wmma_f32_kernel(const float* __restrict__ x, const float* __restrict__ W,
                     float* __restrict__ h) {
    const int lane  = threadIdx.x & 31;
    const int wave  = threadIdx.x >> 5;    // 0..7
    const int m0    = blockIdx.x * 16;
    const int n0    = wave * 16;
    const int half  = lane >> 4;           // 0 or 1
    const int idx16 = lane & 15;           // M for A, N for B

    const float* xrow = x + (size_t)(m0 + idx16) * IN_CH;
    const float* wcol = W + n0 + idx16;

    v8f c = {0.f, 0.f, 0.f, 0.f, 0.f, 0.f, 0.f, 0.f};

#pragma unroll 8
    for (int kk = 0; kk < IN_CH; kk += 4) {
        const int k = kk + 2 * half;
        v2f a = *(const v2f*)(xrow + k);          // A: K = k, k+1 (8B aligned)
        v2f b;
        b.x = wcol[(size_t)k * OUT_CH];           // B: K = k
        b.y = wcol[(size_t)(k + 1) * OUT_CH];     // B: K = k+1
        c = __builtin_amdgcn_wmma_f32_16x16x4_f32(
                /*neg_a=*/false, a, /*neg_b=*/false, b,
                /*c_mod=*/(short)0, c, /*reuse_a=*/false, /*reuse_b=*/false);
    }

    float* hout = h + (size_t)(m0 + 8 * half) * OUT_CH + n0 + idx16;
#pragma unroll
    for (int r = 0; r < 8; ++r) {
        hout[(size_t)r * OUT_CH] = c[r];          // row m0 + 8*half + r
    }
}

// ---------------------------------------------------------------------------
// 5) accumulator init with self-loop term: acc[i][c] = h[i][c] * dinv[i]^2
//    float4-vectorized: 32 float4s per node.
__global__ void acc_init_kernel(const float4* __restrict__ h4,
                                const float* __restrict__ dinv,
                                float4* __restrict__ acc4) {
    int t = blockIdx.x * blockDim.x + threadIdx.x;   // float4 index
    if (t < N_NODES * (OUT_CH / 4)) {
        int i = t >> 5;                               // node id (t / 32)
        float di = dinv[i];
        float nn = di * di;
        float4 v = h4[t];
        v.x *= nn; v.y *= nn; v.z *= nn; v.w *= nn;
        acc4[t] = v;
    }
}

// 6) edge scatter: acc[dst][c] += h[src][c] * dinv[src] * dinv[dst]
//    One wave per edge; lane owns 4 channels (b128 load + 4 f32 atomics).
__global__ void __launch_bounds__(256)
scatter_kernel(const int* __restrict__ edge_index,
               const float* __restrict__ h,
               const float* __restrict__ dinv,
               float* __restrict__ acc) {
    int wid  = (blockIdx.x * blockDim.x + threadIdx.x) >> 5;  // global wave = edge
    int lane = threadIdx.x & 31;
    if (wid < N_EDGES) {
        int s = edge_index[wid];             // edge_index[0][e]
        int d = edge_index[N_EDGES + wid];   // edge_index[1][e]
        float nrm = dinv[s] * dinv[d];
        const float4* hs = (const float4*)(h + (size_t)s * OUT_CH);
        float4 v = hs[lane];                 // channels 4*lane .. 4*lane+3
        float* ad = acc + (size_t)d * OUT_CH + lane * 4;
        __hip_atomic_fetch_add(ad + 0, v.x * nrm,
                               __ATOMIC_RELAXED, __HIP_MEMORY_SCOPE_AGENT);
        __hip_atomic_fetch_add(ad + 1, v.y * nrm,
                               __ATOMIC_RELAXED, __HIP_MEMORY_SCOPE_AGENT);
        __hip_atomic_fetch_add(ad + 2, v.z * nrm,
                               __ATOMIC_RELAXED, __HIP_MEMORY_SCOPE_AGENT);
        __hip_atomic_fetch_add(ad + 3, v.w * nrm,
                               __ATOMIC_RELAXED, __HIP_MEMORY_SCOPE_AGENT);
    }
}

// 7) bias + Leaky neuron: mem_new = BETA*mem + (acc + b) - (mem>TH)*TH
//    spk = mem_new > TH.  acc buffer == mem_new output (in-place rewrite).
//    float4-vectorized.
__global__ void leaky_kernel(const float4* __restrict__ mem4,
                             const float4* __restrict__ bias4,
                             float4* __restrict__ spk4,
                             float4* __restrict__ acc_memnew4) {
    int t = blockIdx.x * blockDim.x + threadIdx.x;   // float4 index
    if (t < N_NODES * (OUT_CH / 4)) {
        int ch4 = t & 31;                             // float4 channel group
        float4 cur = acc_memnew4[t];
        float4 b   = bias4[ch4];
        float4 m   = mem4[t];
        float4 mn, sp;
        mn.x = BETA * m.x + (cur.x + b.x) - ((m.x > THRESH) ? THRESH : 0.0f);
        mn.y = BETA * m.y + (cur.y + b.y) - ((m.y > THRESH) ? THRESH : 0.0f);
        mn.z = BETA * m.z + (cur.z + b.z) - ((m.z > THRESH) ? THRESH : 0.0f);
        mn.w = BETA * m.w + (cur.w + b.w) - ((m.w > THRESH) ? THRESH : 0.0f);
        sp.x = (mn.x > THRESH) ? 1.0f : 0.0f;
        sp.y = (mn.y > THRESH) ? 1.0f : 0.0f;
        sp.z = (mn.z > THRESH) ? 1.0f : 0.0f;
        sp.w = (mn.w > THRESH) ? 1.0f : 0.0f;
        spk4[t] = sp;
        acc_memnew4[t] = mn;
    }
}

// ---------------------------------------------------------------------------
extern "C" void kernel_launch(void* const* d_in, const int* in_sizes, int n_in,
                              void* d_out, int out_size, void* d_ws, size_t ws_size,
                              hipStream_t stream) {
    const float* x    = (const float*)d_in[0];  // [50000, 256]
    const int*   ei   = (const int*)  d_in[1];  // [2, 800000]
    const float* mem  = (const float*)d_in[2];  // [50000, 128]
    const float* W    = (const float*)d_in[3];  // [256, 128]
    const float* bias = (const float*)d_in[4];  // [128]

    // Workspace layout: [ dinv: 50000 f ][ h: 50000*128 f ]
    float* dinv = (float*)d_ws;
    float* h    = (float*)((char*)d_ws + ((sizeof(float) * N_NODES + 255) & ~(size_t)255));

    // Output: [ spk: 6.4M f ][ mem_new: 6.4M f ]; mem_new half doubles as scatter acc.
    float* spk = (float*)d_out;
    float* acc = spk + (size_t)N_NODES * OUT_CH;

    const int B = 256;

    deg_init_kernel<<<(N_NODES + B - 1) / B, B, 0, stream>>>(dinv);
    deg_accum_kernel<<<(N_EDGES + B - 1) / B, B, 0, stream>>>(ei, dinv);
    deg_rsqrt_kernel<<<(N_NODES + B - 1) / B, B, 0, stream>>>(dinv);

    gemm_wmma_f32_kernel<<<N_NODES / 16, 256, 0, stream>>>(x, W, h);

    const int NV4 = N_NODES * (OUT_CH / 4);     // 1.6M float4 elements
    acc_init_kernel<<<(NV4 + B - 1) / B, B, 0, stream>>>(
        (const float4*)h, dinv, (float4*)acc);

    // one wave (32 lanes) per edge -> 8 edges per 256-thread block
    scatter_kernel<<<(N_EDGES + 7) / 8, B, 0, stream>>>(ei, h, dinv, acc);

    leaky_kernel<<<(NV4 + B - 1) / B, B, 0, stream>>>(
        (const float4*)mem, (const float4*)bias, (float4*)spk, (float4*)acc);
}